// BipartiteCommitteeSAGE_48344151884192
// MI455X (gfx1250) — compile-verified
//
#include <hip/hip_runtime.h>

#define N_POL   50000
#define N_TICK  200000
#define N_COMM  5000
#define N_NODES (N_POL + N_TICK + N_COMM)   /* 255000 */
#define N_PAD   255008                      /* next multiple of 16 */
#define N_EDGES 4000000
#define EMB  32
#define HID  64
#define OUTD 32

typedef __attribute__((ext_vector_type(2))) float v2f;
typedef __attribute__((ext_vector_type(8))) float v8f;

static __device__ __forceinline__ v8f wmma_f32_16x16x4(v2f a, v2f b, v8f c) {
    // D = A(16x4) * B(4x16) + C(16x16), full f32, wave32
    return __builtin_amdgcn_wmma_f32_16x16x4_f32(
        /*neg_a=*/false, a, /*neg_b=*/false, b,
        /*c_mod=*/(short)0, c, /*reuse_a=*/false, /*reuse_b=*/false);
}

// ---------------------------------------------------------------------------
// Zero-fill (graph-capture safe replacement for hipMemsetAsync)
// ---------------------------------------------------------------------------
__global__ void zero_f32_kernel(float* __restrict__ p, long long n) {
    long long i = ((long long)blockIdx.x * blockDim.x + threadIdx.x) * 4;
    if (i + 3 < n) {
        *(float4*)(p + i) = make_float4(0.f, 0.f, 0.f, 0.f);
    } else {
        for (; i < n; ++i) p[i] = 0.f;
    }
}

// ---------------------------------------------------------------------------
// Stage 1: fused node embeddings + LayerNorm -> x0 [N_PAD x 32]
// ---------------------------------------------------------------------------
__global__ void node_embed_kernel(
    const float* __restrict__ pol_features,   // [N_POL,7]
    const int*   __restrict__ state_ids,      // [N_POL]
    const int*   __restrict__ sector_ids,     // [N_TICK]
    const int*   __restrict__ industry_ids,   // [N_TICK]
    const float* __restrict__ comp_extra,     // [N_TICK,1]
    const float* __restrict__ pol_W,          // [7,32]
    const float* __restrict__ pol_b,          // [32]
    const float* __restrict__ state_emb,      // [50,32]
    const float* __restrict__ sector_emb,     // [12,8]
    const float* __restrict__ industry_emb,   // [150,8]
    const float* __restrict__ comp_W,         // [17,32]
    const float* __restrict__ comp_b,         // [32]
    const float* __restrict__ comm_emb,       // [N_COMM,32]
    const float* __restrict__ ln_g,           // [32]
    const float* __restrict__ ln_b,           // [32]
    float* __restrict__ x0)                   // [N_PAD,32]
{
    int i = blockIdx.x * blockDim.x + threadIdx.x;
    if (i >= N_PAD) return;

    float v[EMB];
    if (i >= N_NODES) {                       // padding rows: zero
        float* o = x0 + (size_t)i * EMB;
        #pragma unroll
        for (int j = 0; j < EMB; ++j) o[j] = 0.f;
        return;
    }

    if (i < N_POL) {
        float f[7];
        #pragma unroll
        for (int k = 0; k < 7; ++k) f[k] = pol_features[(size_t)i * 7 + k];
        const float* se = state_emb + (size_t)state_ids[i] * EMB;
        #pragma unroll
        for (int j = 0; j < EMB; ++j) {
            float acc = pol_b[j];
            #pragma unroll
            for (int k = 0; k < 7; ++k) acc += f[k] * pol_W[k * EMB + j];
            v[j] = fmaxf(acc, 0.f) + se[j];
        }
    } else if (i < N_POL + N_TICK) {
        int t = i - N_POL;
        float f[17];
        const float* se = sector_emb   + (size_t)sector_ids[t]   * 8;
        const float* ie = industry_emb + (size_t)industry_ids[t] * 8;
        #pragma unroll
        for (int k = 0; k < 8; ++k) { f[k] = se[k]; f[8 + k] = ie[k]; }
        f[16] = comp_extra[t];
        #pragma unroll
        for (int j = 0; j < EMB; ++j) {
            float acc = comp_b[j];
            #pragma unroll
            for (int k = 0; k < 17; ++k) acc += f[k] * comp_W[k * EMB + j];
            v[j] = fmaxf(acc, 0.f);
        }
    } else {
        int c = i - N_POL - N_TICK;
        const float* ce = comm_emb + (size_t)c * EMB;
        #pragma unroll
        for (int j = 0; j < EMB; ++j) v[j] = ce[j];
    }

    // LayerNorm over 32 features
    float mu = 0.f;
    #pragma unroll
    for (int j = 0; j < EMB; ++j) mu += v[j];
    mu *= (1.f / EMB);
    float var = 0.f;
    #pragma unroll
    for (int j = 0; j < EMB; ++j) { float d = v[j] - mu; var += d * d; }
    var *= (1.f / EMB);
    float inv = rsqrtf(var + 1e-5f);

    float* o = x0 + (size_t)i * EMB;
    #pragma unroll
    for (int j = 0; j < EMB; ++j)
        o[j] = (v[j] - mu) * inv * ln_g[j] + ln_b[j];
}

// ---------------------------------------------------------------------------
// Stage 2/4: edge scatter  agg[dst] += x[src] * ew
// One wave covers 32 consecutive features of one edge -> 128B coalesced
// gather + 128B of contiguous f32 atomics (L2-resident working set).
// ---------------------------------------------------------------------------
template<int D>
__global__ void edge_scatter_kernel(
    const int*   __restrict__ src,
    const int*   __restrict__ dst,
    const float* __restrict__ ew,
    const float* __restrict__ x,     // [N_PAD, D]
    float*       __restrict__ agg)   // [N_PAD, D]
{
    constexpr int SHIFT = (D == 32) ? 5 : 6;
    long long tid = (long long)blockIdx.x * blockDim.x + threadIdx.x;
    long long total = (long long)N_EDGES * D;
    if (tid >= total) return;
    int f = (int)(tid & (D - 1));
    long long e = tid >> SHIFT;
    int s = src[e];
    int d = dst[e];
    float w = ew[e];
    atomicAdd(&agg[(size_t)d * D + f], x[(size_t)s * D + f] * w);
}

// ---------------------------------------------------------------------------
// Stage 3/5: dense node update via V_WMMA_F32_16X16X4_F32
//   out = [relu]( agg @ w_rel + xroot @ w_root + bias )
// One wave computes a 16-row x NOUT stripe. ISA operand layouts:
//   A (16x4 f32, 2 VGPRs): lanes 0-15 -> M=lane, K={k,k+1};
//                          lanes 16-31 -> M=lane-16, K={k+2,k+3}
//   B (4x16 f32, 2 VGPRs): same striping, N=lane%16
//   C/D (16x16 f32, 8 VGPRs): VGPR r -> M=r (lanes 0-15) / M=r+8 (lanes 16-31)
// ---------------------------------------------------------------------------
template<int K, int NOUT, bool RELU>
__global__ void dense_gc_kernel(
    const float* __restrict__ agg,     // [N_PAD, K]
    const float* __restrict__ xroot,   // [N_PAD, K]
    const float* __restrict__ w_rel,   // [K, NOUT]
    const float* __restrict__ w_root,  // [K, NOUT]
    const float* __restrict__ bias,    // [NOUT]
    float*       __restrict__ out,     // [n rows, NOUT]
    int n_valid_out)                   // rows actually stored
{
    constexpr int KSTEPS = K / 4;
    constexpr int NTILES = NOUT / 16;
    constexpr int NRB    = N_PAD / 16;

    int wave = threadIdx.x >> 5;
    int lane = threadIdx.x & 31;
    int rb   = blockIdx.x * (blockDim.x >> 5) + wave;
    if (rb >= NRB) return;                       // wave-uniform exit

    int row_base = rb * 16;
    int arow = lane & 15;                        // M within tile
    int col  = lane & 15;                        // N within tile
    int koff = (lane < 16) ? 0 : 2;              // K sub-offset

    // Preload both A operands for all K-steps (reused across N-tiles)
    v2f a_rel[KSTEPS], a_root[KSTEPS];
    const float* ag = agg   + (size_t)(row_base + arow) * K;
    const float* xr = xroot + (size_t)(row_base + arow) * K;
    #pragma unroll
    for (int k = 0; k < KSTEPS; ++k) {
        int ks = k * 4 + koff;
        a_rel[k].x  = ag[ks]; a_rel[k].y  = ag[ks + 1];
        a_root[k].x = xr[ks]; a_root[k].y = xr[ks + 1];
    }

    #pragma unroll
    for (int nt = 0; nt < NTILES; ++nt) {
        v8f acc = {};
        #pragma unroll
        for (int k = 0; k < KSTEPS; ++k) {
            int ks = k * 4 + koff;
            v2f b_rel, b_root;
            b_rel.x  = w_rel [(size_t)ks * NOUT + nt * 16 + col];
            b_rel.y  = w_rel [(size_t)(ks + 1) * NOUT + nt * 16 + col];
            b_root.x = w_root[(size_t)ks * NOUT + nt * 16 + col];
            b_root.y = w_root[(size_t)(ks + 1) * NOUT + nt * 16 + col];
            acc = wmma_f32_16x16x4(a_rel[k],  b_rel,  acc);
            acc = wmma_f32_16x16x4(a_root[k], b_root, acc);
        }
        float bj = bias[nt * 16 + col];
        #pragma unroll
        for (int r = 0; r < 8; ++r) {
            int m = (lane < 16) ? r : (r + 8);
            int grow = row_base + m;
            float v = acc[r] + bj;
            if (RELU) v = fmaxf(v, 0.f);
            if (grow < n_valid_out)
                out[(size_t)grow * NOUT + nt * 16 + col] = v;
        }
    }
}

// ---------------------------------------------------------------------------
extern "C" void kernel_launch(void* const* d_in, const int* in_sizes, int n_in,
                              void* d_out, int out_size, void* d_ws, size_t ws_size,
                              hipStream_t stream) {
    const float* pol_features = (const float*)d_in[0];
    const int*   state_ids    = (const int*)  d_in[1];
    const int*   sector_ids   = (const int*)  d_in[2];
    const int*   industry_ids = (const int*)  d_in[3];
    const float* comp_extra   = (const float*)d_in[4];
    const int*   edge_index   = (const int*)  d_in[5];   // [2, N_EDGES] row-major
    const float* edge_weight  = (const float*)d_in[6];
    const float* pol_W        = (const float*)d_in[7];
    const float* pol_b        = (const float*)d_in[8];
    const float* state_emb    = (const float*)d_in[9];
    const float* sector_emb   = (const float*)d_in[10];
    const float* industry_emb = (const float*)d_in[11];
    const float* comp_W       = (const float*)d_in[12];
    const float* comp_b       = (const float*)d_in[13];
    const float* comm_emb     = (const float*)d_in[14];
    const float* ln_g         = (const float*)d_in[15];
    const float* ln_b         = (const float*)d_in[16];
    const float* w1_rel       = (const float*)d_in[17];
    const float* b1_rel       = (const float*)d_in[18];
    const float* w1_root      = (const float*)d_in[19];
    const float* w2_rel       = (const float*)d_in[20];
    const float* b2_rel       = (const float*)d_in[21];
    const float* w2_root      = (const float*)d_in[22];

    const int* src = edge_index;
    const int* dst = edge_index + N_EDGES;

    // Workspace layout (agg2 aliases x0+agg1, which are dead after dense1):
    //   [ x0 : N_PAD*32 ][ agg1 : N_PAD*32 ][ h1 : N_PAD*64 ]
    //   [            agg2 : N_PAD*64       ]
    float* x0   = (float*)d_ws;
    float* agg1 = x0 + (size_t)N_PAD * EMB;
    float* h1   = agg1 + (size_t)N_PAD * EMB;
    float* agg2 = x0;                             // N_PAD*64 aliased region
    float* outp = (float*)d_out;                  // [N_NODES, 32] f32

    // 1) node embeddings + layernorm
    {
        int threads = 256;
        int blocks = (N_PAD + threads - 1) / threads;
        node_embed_kernel<<<blocks, threads, 0, stream>>>(
            pol_features, state_ids, sector_ids, industry_ids, comp_extra,
            pol_W, pol_b, state_emb, sector_emb, industry_emb,
            comp_W, comp_b, comm_emb, ln_g, ln_b, x0);
    }

    // 2) layer-1 aggregation
    {
        long long n = (long long)N_PAD * EMB;
        zero_f32_kernel<<<(int)((n / 4 + 255) / 256), 256, 0, stream>>>(agg1, n);
        long long total = (long long)N_EDGES * EMB;
        edge_scatter_kernel<EMB><<<(int)((total + 255) / 256), 256, 0, stream>>>(
            src, dst, edge_weight, x0, agg1);
    }

    // 3) layer-1 dense update (WMMA f32): h1 = relu(agg1@w1_rel + x0@w1_root + b1)
    {
        int waves_per_block = 8;                   // 256 threads
        int blocks = (N_PAD / 16 + waves_per_block - 1) / waves_per_block;
        dense_gc_kernel<EMB, HID, true><<<blocks, 256, 0, stream>>>(
            agg1, x0, w1_rel, w1_root, b1_rel, h1, N_PAD);
    }

    // 4) layer-2 aggregation (agg2 overwrites x0/agg1 region — safe: stream-ordered)
    {
        long long n = (long long)N_PAD * HID;
        zero_f32_kernel<<<(int)((n / 4 + 255) / 256), 256, 0, stream>>>(agg2, n);
        long long total = (long long)N_EDGES * HID;
        edge_scatter_kernel<HID><<<(int)((total + 255) / 256), 256, 0, stream>>>(
            src, dst, edge_weight, h1, agg2);
    }

    // 5) layer-2 dense update (WMMA f32): out = agg2@w2_rel + h1@w2_root + b2
    {
        int waves_per_block = 8;
        int blocks = (N_PAD / 16 + waves_per_block - 1) / waves_per_block;
        dense_gc_kernel<HID, OUTD, false><<<blocks, 256, 0, stream>>>(
            agg2, h1, w2_rel, w2_root, b2_rel, outp, N_NODES);
    }
}